// StackedGraphConvLayers_39917426049430
// MI455X (gfx1250) — compile-verified
//
#include <hip/hip_runtime.h>

#define D 128
#define TILE 16

typedef float v2f __attribute__((ext_vector_type(2)));
typedef float v8f __attribute__((ext_vector_type(8)));
typedef int   v4i __attribute__((vector_size(16)));   // matches builtin proto

// --- CDNA5 async global->LDS copy (ASYNCcnt-tracked), with safe fallback ---
#if defined(__has_builtin)
#if __has_builtin(__builtin_amdgcn_global_load_async_to_lds_b128) && \
    __has_builtin(__builtin_amdgcn_s_wait_asynccnt)
#define USE_ASYNC_LDS 1
#endif
#endif
#ifndef USE_ASYNC_LDS
#define USE_ASYNC_LDS 0
#endif

// ---------------------------------------------------------------------------
// Zero the aggregation buffer (float4-vectorized).
// ---------------------------------------------------------------------------
__global__ void zero_kernel(float4* __restrict__ p, int n4) {
    int i = blockIdx.x * blockDim.x + threadIdx.x;
    if (i < n4) p[i] = make_float4(0.f, 0.f, 0.f, 0.f);
}

// ---------------------------------------------------------------------------
// Edge scatter: agg[dst[e]] += h[src[e]]  (sum aggregation).
// One wave32 per edge; each lane handles 4 consecutive floats (float4 load,
// 4x global_atomic_add_f32). Working set is L2-resident (~102MB < 192MB L2),
// so this is bound by L2 atomic throughput, not HBM.
// ---------------------------------------------------------------------------
__global__ void __launch_bounds__(256) scatter_kernel(
    const float* __restrict__ h,
    const int*   __restrict__ src,
    const int*   __restrict__ dst,
    float*       __restrict__ agg,
    int nEdges)
{
    int e = blockIdx.x * 8 + (threadIdx.x >> 5);
    if (e >= nEdges) return;
    int lane = threadIdx.x & 31;
    int s = src[e];
    int d = dst[e];
    const float4 v = *(const float4*)(h + (size_t)s * D + lane * 4);
    float* o = agg + (size_t)d * D + lane * 4;
    atomicAdd(o + 0, v.x);
    atomicAdd(o + 1, v.y);
    atomicAdd(o + 2, v.z);
    atomicAdd(o + 3, v.w);
}

// ---------------------------------------------------------------------------
// Fused GraphConv GEMM:  out = relu(agg @ Wr^T + bias + h @ Wroot^T)
//
// Block = 256 threads = 8 wave32 waves; block owns 16 output rows, wave w
// owns output columns [16w, 16w+16). A-tiles (agg rows, h rows) staged in
// LDS via GLOBAL_LOAD_ASYNC_TO_LDS_B128 (fallback: float4 loads) and shared
// by all 8 waves. K-loop: 32 steps of f32 WMMA 16x16x4, both GEMMs chained
// into a single f32 accumulator (64 WMMAs / tile).
//
// f32 WMMA fragment layout (ISA 7.12.2):
//   A 16x4:  lane l -> row m=l&15, K offsets {koff, koff+1}, koff=2*(l>>4)
//   B 4x16:  lane l -> col n=l&15, same K offsets; B[k][n] = W[n][k]
//   C/D:     VGPR i, lane l -> row i + 8*(l>>4), col l&15
// In-place safe: block only reads its own 16 h-rows (staged before writes).
// ---------------------------------------------------------------------------
__global__ void __launch_bounds__(256) graphconv_gemm_kernel(
    const float* __restrict__ h,      // [N, D] layer input
    const float* __restrict__ agg,    // [N, D] scattered sums
    const float* __restrict__ Wr,     // [D, D] row-major (lin_rel weight)
    const float* __restrict__ bias,   // [D]
    const float* __restrict__ Wroot,  // [D, D] row-major (lin_root weight)
    float*       __restrict__ out,    // [N, D]
    int n)
{
    __shared__ float As[TILE][D];   // agg rows
    __shared__ float Hs[TILE][D];   // h rows

    const int t         = threadIdx.x;
    const int row0      = blockIdx.x * TILE;
    const bool fullTile = (row0 + TILE) <= n;   // block-uniform

    // Stage 16 rows x 128 cols of agg and h into LDS.
    if (fullTile) {
#if USE_ASYNC_LDS
        // CDNA5 async DMA path: 16B per lane per call, no VGPR round-trip.
        for (int f = t; f < TILE * (D / 4); f += 256) {
            int r = f >> 5;            // f / 32
            int c = (f & 31) * 4;
            __builtin_amdgcn_global_load_async_to_lds_b128(
                (v4i*)&agg[(size_t)(row0 + r) * D + c], (v4i*)&As[r][c], 0, 0);
            __builtin_amdgcn_global_load_async_to_lds_b128(
                (v4i*)&h[(size_t)(row0 + r) * D + c],   (v4i*)&Hs[r][c], 0, 0);
        }
        __builtin_amdgcn_s_wait_asynccnt(0);
#else
        for (int f = t; f < TILE * (D / 4); f += 256) {
            int r = f >> 5;
            int c = (f & 31) * 4;
            *(float4*)&As[r][c] = *(const float4*)&agg[(size_t)(row0 + r) * D + c];
            *(float4*)&Hs[r][c] = *(const float4*)&h  [(size_t)(row0 + r) * D + c];
        }
#endif
    } else {
        for (int f = t; f < TILE * (D / 4); f += 256) {
            int r = f >> 5;
            int c = (f & 31) * 4;
            if (row0 + r < n) {
                *(float4*)&As[r][c] = *(const float4*)&agg[(size_t)(row0 + r) * D + c];
                *(float4*)&Hs[r][c] = *(const float4*)&h  [(size_t)(row0 + r) * D + c];
            }
        }
    }
    __syncthreads();

    const int wave = t >> 5;
    const int lane = t & 31;
    const int m    = lane & 15;          // A row / B,D column within tile
    const int koff = (lane >> 4) * 2;    // K sub-offset for this half-wave
    const int n0   = wave * TILE;        // output column tile base

    v8f acc = {};
    #pragma unroll 4
    for (int k0 = 0; k0 < D; k0 += 4) {
        // agg @ Wr^T
        v2f a = *(const v2f*)&As[m][k0 + koff];
        v2f b = *(const v2f*)&Wr[(size_t)(n0 + m) * D + k0 + koff];
        acc = __builtin_amdgcn_wmma_f32_16x16x4_f32(
                  false, a, false, b, (short)0, acc, false, false);
        // h @ Wroot^T
        v2f a2 = *(const v2f*)&Hs[m][k0 + koff];
        v2f b2 = *(const v2f*)&Wroot[(size_t)(n0 + m) * D + k0 + koff];
        acc = __builtin_amdgcn_wmma_f32_16x16x4_f32(
                  false, a2, false, b2, (short)0, acc, false, false);
    }

    // Epilogue: bias + ReLU + store. Guard hoisted to a block-uniform branch
    // so the common path is 8 clean global_store_b32 (no exec-mask churn).
    const float bv = bias[n0 + m];
    float* orow = out + (size_t)(row0 + (lane >> 4) * 8) * D + n0 + m;
    if (fullTile) {
        #pragma unroll
        for (int i = 0; i < 8; ++i) {
            float v = acc[i] + bv;
            orow[(size_t)i * D] = v > 0.f ? v : 0.f;
        }
    } else {
        const int rbase = row0 + (lane >> 4) * 8;
        #pragma unroll
        for (int i = 0; i < 8; ++i) {
            if (rbase + i < n) {
                float v = acc[i] + bv;
                orow[(size_t)i * D] = v > 0.f ? v : 0.f;
            }
        }
    }
}

// ---------------------------------------------------------------------------
// Host launcher: 3 layers, each = zero(agg) -> scatter -> fused GEMM.
// Layer 1: x -> d_out; layers 2,3: in-place on d_out.
// Workspace: only agg (N*D floats = 51.2 MB).
// ---------------------------------------------------------------------------
extern "C" void kernel_launch(void* const* d_in, const int* in_sizes, int n_in,
                              void* d_out, int out_size, void* d_ws, size_t ws_size,
                              hipStream_t stream) {
    const float* x       = (const float*)d_in[0];
    const int*   eidx    = (const int*)  d_in[1];
    const float* W1_rel  = (const float*)d_in[2];
    const float* b1      = (const float*)d_in[3];
    const float* W1_root = (const float*)d_in[4];
    const float* W_rel   = (const float*)d_in[5];
    const float* bb      = (const float*)d_in[6];
    const float* W_root  = (const float*)d_in[7];
    float*       out     = (float*)d_out;

    const int N = in_sizes[0] / D;     // 100000
    const int E = in_sizes[1] / 2;     // 640000
    const int* src = eidx;
    const int* dst = eidx + E;

    float* agg = (float*)d_ws;

    const int n4         = N * D / 4;
    const int zeroBlocks = (n4 + 255) / 256;
    const int scatBlocks = (E + 7) / 8;
    const int gemmBlocks = (N + TILE - 1) / TILE;

    auto layer = [&](const float* hin, const float* Wr, const float* bia,
                     const float* Wro, float* hout) {
        zero_kernel<<<zeroBlocks, 256, 0, stream>>>((float4*)agg, n4);
        scatter_kernel<<<scatBlocks, 256, 0, stream>>>(hin, src, dst, agg, E);
        graphconv_gemm_kernel<<<gemmBlocks, 256, 0, stream>>>(
            hin, agg, Wr, bia, Wro, hout, N);
    };

    layer(x,   W1_rel,        b1,     W1_root,        out);  // layer 1
    layer(out, W_rel,         bb,     W_root,         out);  // layer 2 (in place)
    layer(out, W_rel + D * D, bb + D, W_root + D * D, out);  // layer 3 (in place)
}